// MCTC_35845797052649
// MI455X (gfx1250) — compile-verified
//
#include <hip/hip_runtime.h>
#include <stdint.h>

#define NN   64
#define CHN  40
#define TT   2000
#define LL   256
#define NEGF (-1e30f)

typedef float v2f __attribute__((ext_vector_type(2)));
typedef float v8f __attribute__((ext_vector_type(8)));

// ---- workspace layout (bytes) ----
// losses : 64 f32              @ 0
// u      : 64 i32              @ 256
// tlen   : 64 i32              @ 512
// labels : 64*256 i32          @ 1024
// skeys  : 64*256 u64          @ 66560
// base   : 64*2000 f32         @ 197632
// charp  : 64*256*2000 f32     @ 709632
#define OFF_LOSS   0
#define OFF_U      256
#define OFF_TLEN   512
#define OFF_LABELS 1024
#define OFF_SKEYS  66560
#define OFF_BASE   197632
#define OFF_CHARP  709632

// ------------------------------------------------------------------
// Kernel 1: per-batch preprocessing (np.unique(axis=1) emulation).
// Pack each target column into a 40-bit key (ch0 = MSB -> matches
// np.unique's lexicographic column sort). rank = #distinct smaller keys.
// inv' = (rank+1)%u ; labels = run-collapsed inv' minus first run.
// ------------------------------------------------------------------
__global__ void mctc_preprocess(const int* __restrict__ targets, char* ws) {
    const int n = blockIdx.x;
    const int c = threadIdx.x;                 // column 0..255
    __shared__ unsigned long long kkey[LL];
    __shared__ int isf[LL];
    __shared__ int invv[LL];

    unsigned long long key = 0ull;
    for (int ch = 0; ch < CHN; ++ch) {
        unsigned long long bit = (unsigned long long)(targets[((size_t)n * CHN + ch) * LL + c] & 1);
        key |= bit << (39 - ch);
    }
    kkey[c] = key;

    unsigned long long* skeys = (unsigned long long*)(ws + OFF_SKEYS) + (size_t)n * LL;
    skeys[c] = 0ull;                           // zero-fill before scatter
    __syncthreads();

    int first = 1;
    for (int d = 0; d < c; ++d)
        if (kkey[d] == key) { first = 0; break; }
    isf[c] = first;
    __syncthreads();

    int rank = 0, u = 0;
    for (int d = 0; d < LL; ++d) {
        if (isf[d]) {
            ++u;
            if (kkey[d] < key) ++rank;
        }
    }
    invv[c] = (rank + 1) % u;
    if (first) skeys[rank] = key;              // sorted unique keys
    __syncthreads();

    if (c == 0) {
        int* labn = (int*)(ws + OFF_LABELS) + (size_t)n * LL;
        int prev = invv[0];
        int len = 0;
        for (int j = 1; j < LL; ++j) {
            if (invv[j] != prev) { labn[len++] = invv[j]; prev = invv[j]; }
        }
        for (int j = len; j < LL; ++j) labn[j] = 0;
        ((int*)(ws + OFF_TLEN))[n] = len;
        ((int*)(ws + OFF_U))[n]    = u;
    }
}

// ------------------------------------------------------------------
// Kernel 2: base[n][t] = sum_ch lp0[n][ch][t]
// ------------------------------------------------------------------
__global__ void mctc_base(const float* __restrict__ lp, char* ws) {
    const int n = blockIdx.x;
    const int t = blockIdx.y * blockDim.x + threadIdx.x;
    if (t >= TT) return;
    const float* lp0 = lp + ((size_t)(n * 2 + 0) * CHN) * TT;
    float s = 0.f;
    for (int ch = 0; ch < CHN; ++ch) s += lp0[(size_t)ch * TT + t];
    ((float*)(ws + OFF_BASE))[(size_t)n * TT + t] = s;
}

// ------------------------------------------------------------------
// Kernel 3: char_probs via V_WMMA_F32_16X16X4_F32.
//   charp[n][i][t] = base[n][t] + sum_k bit(skey[i-1],k) * (lp1-lp0)[k][t]
// A (16x4 f32): lanes 0-15 -> {K0,K1}, lanes 16-31 -> {K2,K3}; M = lane&15
// B (4x16 f32): mirrored; N = lane&15
// C/D (16x16 f32): VGPR r -> M = r (lanes 0-15) / r+8 (lanes 16-31)
//
// B fragments are loaded ONCE per wave (they don't depend on the M-tile)
// and reused for both M-tiles. A bits come from a constant-shift select
// on the pre-shifted key (32-bit-cheap, no u64->f32 convert chain).
// ------------------------------------------------------------------
__global__ void mctc_wmma(const float* __restrict__ lp, char* ws) {
    const int n    = blockIdx.y;
    const int t0   = blockIdx.x * 16;          // 125 tiles * 16 = 2000
    const int wave = threadIdx.x >> 5;         // 8 waves
    const int lane = threadIdx.x & 31;

    const unsigned long long* skeys = (const unsigned long long*)(ws + OFF_SKEYS) + (size_t)n * LL;
    const float* basev = (const float*)(ws + OFF_BASE) + (size_t)n * TT;
    float* charp = (float*)(ws + OFF_CHARP) + (size_t)n * LL * TT;

    const float* d0 = lp + ((size_t)(n * 2 + 0) * CHN) * TT;
    const float* d1 = lp + ((size_t)(n * 2 + 1) * CHN) * TT;

    const int tcol  = t0 + (lane & 15);
    const int khalf = (lane >> 4) * 2;         // 0 or 2

    // ---- load B fragments once (independent of M-tile) ----
    v2f bfrag[CHN / 4];
    #pragma unroll
    for (int i = 0; i < CHN / 4; ++i) {
        const int ka = i * 4 + khalf;
        if (i + 1 < CHN / 4) {                 // global_prefetch_b8 on B stream
            __builtin_prefetch(&d1[(size_t)(ka + 4) * TT + tcol], 0, 1);
            __builtin_prefetch(&d0[(size_t)(ka + 4) * TT + tcol], 0, 1);
        }
        bfrag[i][0] = d1[(size_t)(ka + 0) * TT + tcol] - d0[(size_t)(ka + 0) * TT + tcol];
        bfrag[i][1] = d1[(size_t)(ka + 1) * TT + tcol] - d0[(size_t)(ka + 1) * TT + tcol];
    }
    const float bb = basev[tcol];

    for (int mt = wave; mt < 16; mt += 8) {    // 16 M-tiles, 2 per wave
        const int m0   = mt * 16;
        const int mrow = m0 + (lane & 15);     // char row for this A lane
        // char row i>=1 uses sorted key i-1 ; row 0 gets zero bits
        const unsigned long long key = (mrow == 0) ? 0ull : skeys[mrow - 1];
        // pre-shift so each unrolled step uses a compile-time-constant shift
        const unsigned long long keyAdj = key << khalf;

        v8f acc = {};
        #pragma unroll
        for (int i = 0; i < CHN / 4; ++i) {
            const int ko = i * 4;
            v2f a;
            a[0] = ((keyAdj >> (39 - ko)) & 1ull) ? 1.0f : 0.0f;
            a[1] = ((keyAdj >> (38 - ko)) & 1ull) ? 1.0f : 0.0f;
            acc = __builtin_amdgcn_wmma_f32_16x16x4_f32(
                /*neg_a=*/false, a, /*neg_b=*/false, bfrag[i],
                /*c_mod=*/(short)0, acc, /*reuse_a=*/false, /*reuse_b=*/false);
        }

        const int mbase = m0 + ((lane >> 4) ? 8 : 0);
        #pragma unroll
        for (int r = 0; r < 8; ++r) {
            charp[(size_t)(mbase + r) * TT + tcol] = acc[r] + bb;
        }
    }
}

// ------------------------------------------------------------------
// Kernel 4: CTC alpha scan, one block per batch, state-per-thread,
// alpha double-buffered in LDS. Prologue overwrites charp row 0 with
// the blank prob lp1[ch0] (row 0 from the WMMA kernel is base-only).
// ------------------------------------------------------------------
__global__ void mctc_scan(const float* __restrict__ lp, char* ws) {
    const int n   = blockIdx.x;
    const int s   = threadIdx.x;               // state 0..511
    float* cp = (float*)(ws + OFF_CHARP) + (size_t)n * LL * TT;

    // fixup: blank row = lp1[n][0][:]
    const float* lp1c0 = lp + ((size_t)(n * 2 + 1) * CHN + 0) * TT;
    for (int t = s; t < TT; t += 512) cp[t] = lp1c0[t];
    __syncthreads();

    const int S  = ((const int*)(ws + OFF_TLEN))[n];
    const int ns = 2 * S + 1;                  // <= 511
    const int* labn = (const int*)(ws + OFF_LABELS) + (size_t)n * LL;

    int myidx = 0, myskip = 0;
    if (s < ns && (s & 1)) {
        const int j = (s - 1) >> 1;
        const int lab = labn[j];
        myidx = lab;
        if (s >= 3) {
            const int labp = labn[j - 1];
            myskip = (lab != labp) && (lab != 0);
        }
    }
    const float* myrow = cp + (size_t)myidx * TT;

    __shared__ float bufA[512];
    __shared__ float bufB[512];
    bufA[s] = (s < ns && s < 2) ? myrow[0] : NEGF;
    __syncthreads();

    float* cur = bufA;
    float* nxt = bufB;
    for (int t = 1; t < TT; ++t) {
        float v = NEGF;
        if (s < ns) {
            const float a  = cur[s];
            const float a1 = (s >= 1) ? cur[s - 1] : NEGF;
            const float a2 = myskip   ? cur[s - 2] : NEGF;
            const float m  = fmaxf(a, fmaxf(a1, a2));
            const float tot = m + logf(expf(a - m) + expf(a1 - m) + expf(a2 - m));
            v = tot + myrow[t];
        }
        nxt[s] = v;
        __syncthreads();
        float* tmp = cur; cur = nxt; nxt = tmp;
    }

    if (s == 0) {
        const float ea = cur[2 * S];
        const float eb = cur[2 * S - 1];
        const float m  = fmaxf(ea, eb);
        ((float*)(ws + OFF_LOSS))[n] = -(m + logf(expf(ea - m) + expf(eb - m)));
    }
}

// ------------------------------------------------------------------
// Kernel 5: mean of 64 losses
// ------------------------------------------------------------------
__global__ void mctc_reduce(const char* ws, float* out) {
    __shared__ float sh[NN];
    const int i = threadIdx.x;
    sh[i] = ((const float*)(ws + OFF_LOSS))[i];
    __syncthreads();
    if (i == 0) {
        float s = 0.f;
        for (int j = 0; j < NN; ++j) s += sh[j];
        out[0] = s / (float)NN;
    }
}

extern "C" void kernel_launch(void* const* d_in, const int* in_sizes, int n_in,
                              void* d_out, int out_size, void* d_ws, size_t ws_size,
                              hipStream_t stream) {
    (void)in_sizes; (void)n_in; (void)out_size; (void)ws_size;
    const float* lp      = (const float*)d_in[0];   // (64, 2, 40, 2000) f32
    const int*   targets = (const int*)d_in[1];     // (64, 40, 256) i32
    char* ws = (char*)d_ws;
    float* out = (float*)d_out;

    mctc_preprocess<<<dim3(NN), dim3(LL), 0, stream>>>(targets, ws);
    mctc_base<<<dim3(NN, (TT + 255) / 256), dim3(256), 0, stream>>>(lp, ws);
    mctc_wmma<<<dim3(TT / 16, NN), dim3(256), 0, stream>>>(lp, ws);
    mctc_scan<<<dim3(NN), dim3(512), 0, stream>>>(lp, ws);
    mctc_reduce<<<dim3(1), dim3(NN), 0, stream>>>(ws, out);
}